// PointPillar_21603685499734
// MI455X (gfx1250) — compile-verified
//
#include <hip/hip_runtime.h>
#include <stdint.h>

#define K_TOP 4096
#define THREADS 256
#define TILE_BOXES 1024
#define TILE_F4 768          // TILE_BOXES * 3 floats / 4 per float4
#define NBINS 32768
#define CAND_CAP 8192
#define C2_CAP 131072

// ---- CDNA5 async global->LDS path (guarded; falls back to vector loads) ----
#if defined(__gfx1250__) && defined(__HIP_DEVICE_COMPILE__) && \
    __has_builtin(__builtin_amdgcn_global_load_async_to_lds_b128) && \
    __has_builtin(__builtin_amdgcn_s_wait_asynccnt)
#define USE_ASYNC_LDS 1
#else
#define USE_ASYNC_LDS 0
#endif

#if USE_ASYNC_LDS
typedef int v4i __attribute__((vector_size(16)));
typedef __attribute__((address_space(1))) v4i global_v4i;
typedef __attribute__((address_space(3))) v4i lds_v4i;
__device__ __forceinline__ global_v4i* to_global_v4(const void* p) {
  return (global_v4i*)(unsigned long long)(uintptr_t)p;
}
__device__ __forceinline__ lds_v4i* to_lds_v4(const void* p) {
  // generic LDS pointer: low 32 bits are the LDS byte offset (ISA 10.2 aperture rules)
  return (lds_v4i*)(unsigned)(uintptr_t)p;
}
#endif

__device__ __forceinline__ unsigned long long comp_entry(unsigned key,
                                                         unsigned idx) {
  // sort descending => equal scores order by ascending index via ~idx
  return ((unsigned long long)key << 32) | (unsigned)(~idx);
}

// ---------------- Pass 1: score = max over 3 classes, key + histogram -------
// Double-buffered async global->LDS pipeline: DMA for tile t+1 overlaps the
// VALU work on tile t; s_wait_asynccnt(3) retires exactly the older tile's
// 3 per-thread B128 async copies (async loads complete in issue order).
__global__ __launch_bounds__(THREADS) void k_score_hist(
    const float* __restrict__ cls, unsigned* __restrict__ keys,
    unsigned* __restrict__ hist1, long nBoxes, long numTiles, long totalF4) {
  __shared__ float4 tile[2][TILE_F4];  // 2 x 12 KB
  const int tid = threadIdx.x;

#if USE_ASYNC_LDS
#define ISSUE_TILE(T_IDX, BUF)                                              \
  {                                                                         \
    long f4b = (T_IDX) * (long)TILE_F4;                                     \
    _Pragma("unroll") for (int u = 0; u < 3; ++u) {                         \
      long i = tid + u * THREADS;                                           \
      long src = f4b + i;                                                   \
      if (src >= totalF4) src = totalF4 - 1; /* clamp tail: stay in-bounds */\
      __builtin_amdgcn_global_load_async_to_lds_b128(                       \
          to_global_v4(((const float4*)cls) + src),                         \
          to_lds_v4(&tile[BUF][i]), 0, 0);                                  \
    }                                                                       \
  }
#else
#define ISSUE_TILE(T_IDX, BUF)                                              \
  {                                                                         \
    long f4b = (T_IDX) * (long)TILE_F4;                                     \
    _Pragma("unroll") for (int u = 0; u < 3; ++u) {                         \
      long i = tid + u * THREADS;                                           \
      long src = f4b + i;                                                   \
      if (src >= totalF4) src = totalF4 - 1;                                \
      tile[BUF][i] = ((const float4*)cls)[src];                             \
    }                                                                       \
  }
#endif

  long t = blockIdx.x;
  int cur = 0;
  if (t < numTiles) ISSUE_TILE(t, 0);
  for (; t < numTiles; t += gridDim.x) {
    long tn = t + gridDim.x;
    bool hasNext = tn < numTiles;
    if (hasNext) ISSUE_TILE(tn, 1 - cur);
#if USE_ASYNC_LDS
    if (hasNext)
      __builtin_amdgcn_s_wait_asynccnt(3);  // only tile t's copies retired
    else
      __builtin_amdgcn_s_wait_asynccnt(0);
#endif
    __syncthreads();

    long box0 = t * (long)TILE_BOXES + (long)tid * 4;
    if (box0 < nBoxes) {
      float4 a = tile[cur][tid * 3 + 0];
      float4 b = tile[cur][tid * 3 + 1];
      float4 c = tile[cur][tid * 3 + 2];
      float sc[4];
      sc[0] = fmaxf(a.x, fmaxf(a.y, a.z));
      sc[1] = fmaxf(a.w, fmaxf(b.x, b.y));
      sc[2] = fmaxf(b.z, fmaxf(b.w, c.x));
      sc[3] = fmaxf(c.y, fmaxf(c.z, c.w));
#pragma unroll
      for (int u = 0; u < 4; ++u) {
        long bi = box0 + u;
        if (bi < nBoxes) {
          unsigned key = __float_as_uint(sc[u]);  // scores >= 0: bits monotonic
          keys[bi] = key;
          atomicAdd(&hist1[key >> 15], 1u);
        }
      }
    }
    cur ^= 1;
    __syncthreads();  // everyone done reading before this buffer is re-filled
  }
#undef ISSUE_TILE
}

// ------------- Threshold finder over a 32768-bin histogram ------------------
// phase 0: hist1 -> ctl[0]=B, ctl[1]=count strictly above bin B
// phase 1: hist2 -> ctl[2]=T(key), ctl[3]=count strictly above T, ctl[4]=#ties
__global__ __launch_bounds__(1024) void k_find_threshold(
    const unsigned* __restrict__ hist, unsigned* __restrict__ ctl, int phase) {
  __shared__ unsigned sums[1024];
  const int tid = threadIdx.x;
  const int per = NBINS / 1024;  // 32 bins per thread
  unsigned s = 0;
  int base = tid * per;
  for (int i = 0; i < per; ++i) s += hist[base + i];
  sums[tid] = s;
  __syncthreads();
  if (tid == 0) {
    unsigned K = (phase == 0) ? (unsigned)K_TOP : ((unsigned)K_TOP - ctl[1]);
    unsigned above = 0;
    int chunk = 0;
    for (int c = 1023; c >= 0; --c) {
      if (above + sums[c] >= K) { chunk = c; break; }
      above += sums[c];
    }
    int binSel = chunk * per;
    for (int b = chunk * per + per - 1; b >= chunk * per; --b) {
      unsigned h = hist[b];
      if (above + h >= K) { binSel = b; break; }
      above += h;
    }
    if (phase == 0) {
      ctl[0] = (unsigned)binSel;
      ctl[1] = above;
    } else {
      unsigned aboveT = ctl[1] + above;
      ctl[2] = (ctl[0] << 15) | (unsigned)binSel;
      ctl[3] = aboveT;
      ctl[4] = (unsigned)K_TOP - aboveT;
    }
  }
}

// ------ Pass 3: bins>B -> cand directly; bin==B -> spill to C2 + hist2 ------
__device__ __forceinline__ void filter_one(unsigned k, unsigned idx, unsigned B,
                                           unsigned* hist2,
                                           unsigned long long* cand,
                                           unsigned long long* c2,
                                           unsigned* counters) {
  unsigned bin = k >> 15;
  if (bin >= B) {
    if (bin > B) {  // strictly above threshold bin => definitely in top-k set
      unsigned pos = atomicAdd(&counters[0], 1u);
      if (pos < CAND_CAP) cand[pos] = comp_entry(k, idx);
    } else {
      unsigned pos = atomicAdd(&counters[2], 1u);
      if (pos < C2_CAP) c2[pos] = comp_entry(k, idx);
      atomicAdd(&hist2[k & 0x7FFFu], 1u);
    }
  }
}

__global__ __launch_bounds__(THREADS) void k_filter(
    const unsigned* __restrict__ keys, unsigned* __restrict__ hist2,
    const unsigned* __restrict__ ctl, unsigned long long* __restrict__ cand,
    unsigned long long* __restrict__ c2, unsigned* __restrict__ counters,
    long n) {
  unsigned B = ctl[0];
  long n4 = n >> 2;
  long gid = (long)blockIdx.x * blockDim.x + threadIdx.x;
  long stride = (long)gridDim.x * blockDim.x;
  const uint4* k4 = (const uint4*)keys;
  for (long i = gid; i < n4; i += stride) {
    uint4 kv = k4[i];
    unsigned b0 = (unsigned)(i << 2);
    filter_one(kv.x, b0 + 0, B, hist2, cand, c2, counters);
    filter_one(kv.y, b0 + 1, B, hist2, cand, c2, counters);
    filter_one(kv.z, b0 + 2, B, hist2, cand, c2, counters);
    filter_one(kv.w, b0 + 3, B, hist2, cand, c2, counters);
  }
  long tail0 = n4 << 2;
  long rem = n - tail0;
  if (gid < rem)
    filter_one(keys[tail0 + gid], (unsigned)(tail0 + gid), B, hist2, cand, c2,
               counters);
}

// ------------- Pass 5: refine C2 (~bin-B spill) against exact T -------------
__global__ __launch_bounds__(THREADS) void k_refine(
    const unsigned long long* __restrict__ c2, const unsigned* __restrict__ ctl,
    unsigned long long* __restrict__ cand, unsigned* __restrict__ eq,
    unsigned* __restrict__ counters) {
  unsigned T = ctl[2];
  unsigned m = counters[2] < (unsigned)C2_CAP ? counters[2] : (unsigned)C2_CAP;
  unsigned gid = blockIdx.x * blockDim.x + threadIdx.x;
  unsigned stride = gridDim.x * blockDim.x;
  for (unsigned i = gid; i < m; i += stride) {
    unsigned long long e = c2[i];
    unsigned k = (unsigned)(e >> 32);
    if (k > T) {
      unsigned pos = atomicAdd(&counters[0], 1u);
      if (pos < CAND_CAP) cand[pos] = e;
    } else if (k == T) {
      unsigned pos = atomicAdd(&counters[1], 1u);
      if (pos < (unsigned)K_TOP) eq[pos] = ~(unsigned)e;  // raw box index
    }
  }
}

// ------------- Final: tie-select, bitonic sort 4096, write outputs ----------
__global__ __launch_bounds__(1024) void k_final(
    const unsigned long long* __restrict__ cand, const unsigned* __restrict__ eq,
    const unsigned* __restrict__ ctl, const float* __restrict__ cls,
    const float* __restrict__ box, float* __restrict__ outScores,
    float* __restrict__ outBoxes, float* __restrict__ outLabels, long nBoxes) {
  __shared__ unsigned long long s64[K_TOP];  // 32 KB
  unsigned* s32 = (unsigned*)s64;
  const int tid = threadIdx.x;
  unsigned T = ctl[2];
  unsigned K1 = ctl[3] < (unsigned)K_TOP ? ctl[3] : (unsigned)K_TOP;
  unsigned eqN = ctl[9] < (unsigned)K_TOP ? ctl[9] : (unsigned)K_TOP;

  // Sort tie indices ascending (top_k takes lowest indices first on ties).
  for (int i = tid; i < K_TOP; i += 1024)
    s32[i] = (i < (int)eqN) ? eq[i] : 0xFFFFFFFFu;
  __syncthreads();
  for (int k = 2; k <= K_TOP; k <<= 1) {
    for (int j = k >> 1; j > 0; j >>= 1) {
      for (int t = tid; t < K_TOP / 2; t += 1024) {
        int i = ((t & ~(j - 1)) << 1) | (t & (j - 1));
        int p = i | j;
        unsigned a = s32[i], b = s32[p];
        bool up = ((i & k) == 0);
        if (up ? (a > b) : (a < b)) { s32[i] = b; s32[p] = a; }
      }
      __syncthreads();
    }
  }
  // Build 4096 composite entries.
  unsigned long long myComp[4];
#pragma unroll
  for (int u = 0; u < 4; ++u) {
    int s = tid + u * 1024;
    if (s >= (int)K1) {
      unsigned idx = s32[s - (int)K1];
      myComp[u] = ((unsigned long long)T << 32) | (unsigned)(~idx);
    }
  }
  __syncthreads();
#pragma unroll
  for (int u = 0; u < 4; ++u) {
    int s = tid + u * 1024;
    if (s < (int)K1) myComp[u] = cand[s];
    s64[s] = myComp[u];
  }
  __syncthreads();
  for (int k = 2; k <= K_TOP; k <<= 1) {
    for (int j = k >> 1; j > 0; j >>= 1) {
      for (int t = tid; t < K_TOP / 2; t += 1024) {
        int i = ((t & ~(j - 1)) << 1) | (t & (j - 1));
        int p = i | j;
        unsigned long long a = s64[i], b = s64[p];
        bool up = ((i & k) == 0);
        if (up ? (a < b) : (a > b)) { s64[i] = b; s64[p] = a; }  // descending
      }
      __syncthreads();
    }
  }
  // Emit: scores, gathered boxes, labels (argmax+1, first-max wins like jnp.argmax)
#pragma unroll
  for (int u = 0; u < 4; ++u) {
    int r = tid + u * 1024;
    unsigned long long comp = s64[r];
    unsigned key = (unsigned)(comp >> 32);
    unsigned idx = ~(unsigned)comp;
    if (idx >= (unsigned)nBoxes) idx = 0;  // safety clamp
    outScores[r] = __uint_as_float(key);
    const float* bp = box + (size_t)idx * 7;
    float* ob = outBoxes + (size_t)r * 7;
#pragma unroll
    for (int q = 0; q < 7; ++q) ob[q] = bp[q];
    const float* cp = cls + (size_t)idx * 3;
    float c0 = cp[0], c1 = cp[1], c2 = cp[2];
    int lab = 0;
    float bv = c0;
    if (c1 > bv) { bv = c1; lab = 1; }
    if (c2 > bv) { bv = c2; lab = 2; }
    outLabels[r] = (float)(lab + 1);
  }
}

extern "C" void kernel_launch(void* const* d_in, const int* in_sizes, int n_in,
                              void* d_out, int out_size, void* d_ws, size_t ws_size,
                              hipStream_t stream) {
  const float* cls = (const float*)d_in[0];
  const float* box = (const float*)d_in[1];
  long nBoxes = (long)in_sizes[0] / 3;

  char* ws = (char*)d_ws;
  unsigned* hist1 = (unsigned*)(ws);                 // 128 KB
  unsigned* hist2 = (unsigned*)(ws + (128 << 10));   // 128 KB
  unsigned* ctl = (unsigned*)(ws + (256 << 10));     // ctl[0..7], counters [8..10]
  unsigned long long* cand = (unsigned long long*)(ws + (512 << 10));  // 64 KB
  unsigned* eq = (unsigned*)(ws + (768 << 10));      // 16 KB
  unsigned long long* c2 = (unsigned long long*)(ws + (1 << 20));  // 1 MB
  unsigned* keys = (unsigned*)(ws + (2 << 20));      // nBoxes * 4B

  // Deterministic per-call: re-zero histograms/control/counters every launch.
  (void)hipMemsetAsync(d_ws, 0, (1 << 20), stream);

  long totalF4 = (nBoxes * 3 + 3) / 4;
  long numTiles = (nBoxes + TILE_BOXES - 1) / TILE_BOXES;
  int blocks1 = (int)(numTiles < 4096 ? numTiles : 4096);

  k_score_hist<<<blocks1, THREADS, 0, stream>>>(cls, keys, hist1, nBoxes,
                                                numTiles, totalF4);
  k_find_threshold<<<1, 1024, 0, stream>>>(hist1, ctl, 0);
  k_filter<<<2048, THREADS, 0, stream>>>(keys, hist2, ctl, cand, c2, ctl + 8,
                                         nBoxes);
  k_find_threshold<<<1, 1024, 0, stream>>>(hist2, ctl, 1);
  k_refine<<<64, THREADS, 0, stream>>>(c2, ctl, cand, eq, ctl + 8);

  float* outScores = (float*)d_out;
  float* outBoxes = outScores + K_TOP;
  float* outLabels = outBoxes + (size_t)K_TOP * 7;
  k_final<<<1, 1024, 0, stream>>>(cand, eq, ctl, cls, box, outScores, outBoxes,
                                  outLabels, nBoxes);
}